// GCNEncoder_42640435314985
// MI455X (gfx1250) — compile-verified
//
#include <hip/hip_runtime.h>

typedef __attribute__((ext_vector_type(2))) float v2f;
typedef __attribute__((ext_vector_type(8))) float v8f;

#define ATOMIC_FADD(p, v) \
  __hip_atomic_fetch_add((p), (v), __ATOMIC_RELAXED, __HIP_MEMORY_SCOPE_AGENT)

// ---------------- degree / normalization ----------------

__global__ void dinv_init_kernel(float* __restrict__ dinv, int N) {
    int i = blockIdx.x * blockDim.x + threadIdx.x;
    if (i < N) dinv[i] = 1.0f;               // self-loop contributes 1 to degree
}

__global__ void deg_accum_kernel(const int* __restrict__ col,
                                 float* __restrict__ deg, int E) {
    int e = blockIdx.x * blockDim.x + threadIdx.x;
    if (e < E) ATOMIC_FADD(&deg[col[e]], 1.0f);
}

__global__ void deg_finish_kernel(float* __restrict__ dinv, int N) {
    int i = blockIdx.x * blockDim.x + threadIdx.x;
    if (i < N) dinv[i] = rsqrtf(dinv[i]);    // deg >= 1 always (self-loop)
}

// ---------------- dense transform: Out[N,128] = A[N,128] @ W[128,128] ----------------
// One wave per 16x16 output tile. K-loop of 32 x V_WMMA_F32_16X16X4_F32.
// A-frag (16x4 f32): lanes 0-15 hold M=lane, K={0,1}; lanes 16-31 hold M=lane-16, K={2,3}.
// B-frag (4x16 f32): lanes 0-15 hold N=lane, K={0,1}; lanes 16-31 hold N=lane-16, K={2,3}.
// C/D (16x16 f32):   VGPR v: lanes 0-15 -> M=v, N=lane; lanes 16-31 -> M=v+8, N=lane-16.
__global__ __launch_bounds__(256)
void gemm_wmma_f32_kernel(const float* __restrict__ A, int lda,
                          const float* __restrict__ W,   // [128,128] row-major
                          float* __restrict__ Out,       // [N,128] contiguous
                          int N) {
    const int lane   = threadIdx.x & 31;
    const int wave   = threadIdx.x >> 5;     // 0..7 : 16-wide column tile
    const int rtile  = blockIdx.x;           // 16-row tile
    const int halfK  = (lane >> 4) * 2;      // 0 for lanes 0-15, 2 for lanes 16-31
    const int mLane  = rtile * 16 + (lane & 15);
    const int nLane  = wave * 16 + (lane & 15);

    // clamp (not predicate) so EXEC stays all-ones at the WMMA
    const int mClamped = mLane < N ? mLane : N - 1;
    const float* __restrict__ arow = A + (size_t)mClamped * lda;
    const float* __restrict__ bcol = W + nLane;

    v8f acc = {};
    #pragma unroll 4
    for (int k = 0; k < 128; k += 4) {
        v2f a, b;
        a.x = arow[k + halfK];
        a.y = arow[k + halfK + 1];
        b.x = bcol[(k + halfK) * 128];
        b.y = bcol[(k + halfK + 1) * 128];
        acc = __builtin_amdgcn_wmma_f32_16x16x4_f32(
            /*neg_a=*/false, a, /*neg_b=*/false, b,
            /*c_mod=*/(short)0, acc, /*reuse_a=*/false, /*reuse_b=*/false);
    }

    // Store D. Fast path: whole 16-row tile in range (wave-uniform test, no
    // per-lane exec juggling). N=100000 is a multiple of 16, so the guarded
    // tail never runs here, but keep it for generality.
    const int rowBase = rtile * 16 + ((lane >> 4) * 8);
    float* __restrict__ dst = Out + (size_t)rowBase * 128 + nLane;
    if (rtile * 16 + 16 <= N) {
        #pragma unroll
        for (int v = 0; v < 8; ++v) dst[(size_t)v * 128] = acc[v];
    } else {
        #pragma unroll
        for (int v = 0; v < 8; ++v)
            if (rowBase + v < N) dst[(size_t)v * 128] = acc[v];
    }
}

// ---------------- self-loop term: plain store initializes the output block ----------------
// Out[n, off + j] = H[n, j] * dinv[n]^2   (one wave per node, float4 per lane)
__global__ __launch_bounds__(256)
void self_loop_kernel(const float* __restrict__ H,
                      const float* __restrict__ dinv,
                      float* __restrict__ Out, int colOff, int N) {
    int gid  = blockIdx.x * blockDim.x + threadIdx.x;
    int n    = gid >> 5;
    int lane = gid & 31;
    if (n >= N) return;
    float w  = dinv[n];
    w = w * w;
    float4 h = ((const float4*)(H + (size_t)n * 128))[lane];
    float4 o;
    o.x = h.x * w; o.y = h.y * w; o.z = h.z * w; o.w = h.w * w;
    ((float4*)(Out + (size_t)n * 256 + colOff))[lane] = o;
}

// ---------------- edge scatter: Out[col, off+j] += H[row, j] * dinv[row]*dinv[col] ----------------
// One wave per edge; float4 gather (L2-resident), 4 hardware fadd atomics per lane.
__global__ __launch_bounds__(256)
void edge_agg_kernel(const int* __restrict__ rowIdx,
                     const int* __restrict__ colIdx,
                     const float* __restrict__ dinv,
                     const float* __restrict__ H,
                     float* __restrict__ Out, int colOff, int E) {
    int gid  = blockIdx.x * blockDim.x + threadIdx.x;
    int e    = gid >> 5;
    int lane = gid & 31;
    if (e >= E) return;
    int r = rowIdx[e];
    int c = colIdx[e];
    float w = dinv[r] * dinv[c];
    float4 h = ((const float4*)(H + (size_t)r * 128))[lane];
    float* dst = Out + (size_t)c * 256 + colOff + lane * 4;
    ATOMIC_FADD(dst + 0, h.x * w);
    ATOMIC_FADD(dst + 1, h.y * w);
    ATOMIC_FADD(dst + 2, h.z * w);
    ATOMIC_FADD(dst + 3, h.w * w);
}

// ---------------- bias + relu, in place on the output block ----------------
__global__ __launch_bounds__(256)
void bias_relu_kernel(float* __restrict__ Out, const float* __restrict__ b,
                      int colOff, int N) {
    int gid = blockIdx.x * blockDim.x + threadIdx.x;
    int n = gid >> 7;
    int j = gid & 127;
    if (n >= N) return;
    float* p = Out + (size_t)n * 256 + colOff + j;
    float v = *p + b[j];
    *p = v > 0.0f ? v : 0.0f;
}

// ---------------- host orchestration ----------------

static inline int cdiv(int a, int b) { return (a + b - 1) / b; }

extern "C" void kernel_launch(void* const* d_in, const int* in_sizes, int n_in,
                              void* d_out, int out_size, void* d_ws, size_t ws_size,
                              hipStream_t stream) {
    const float* x  = (const float*)d_in[0];
    const int*   ei = (const int*)d_in[1];
    const float* W1 = (const float*)d_in[2];
    const float* b1 = (const float*)d_in[3];
    const float* W2 = (const float*)d_in[4];
    const float* b2 = (const float*)d_in[5];

    const int N = in_sizes[0] / 128;
    const int E = in_sizes[1] / 2;
    const int* rowIdx = ei;        // edge_index[0]
    const int* colIdx = ei + E;    // edge_index[1]

    float* out  = (float*)d_out;             // [N, 256]
    float* dinv = (float*)d_ws;              // [N]
    float* H0   = dinv + ((N + 63) & ~63);   // [N, 128] transformed features

    const int B = 256;

    // symmetric normalization coefficients
    dinv_init_kernel<<<cdiv(N, B), B, 0, stream>>>(dinv, N);
    deg_accum_kernel<<<cdiv(E, B), B, 0, stream>>>(colIdx, dinv, E);
    deg_finish_kernel<<<cdiv(N, B), B, 0, stream>>>(dinv, N);

    const int gemmGrid = cdiv(N, 16);

    // ---- layer 1: h1 = relu(agg(x @ W1) + b1) -> d_out cols [0,128)
    gemm_wmma_f32_kernel<<<gemmGrid, 256, 0, stream>>>(x, 128, W1, H0, N);
    self_loop_kernel<<<cdiv(N * 32, B), B, 0, stream>>>(H0, dinv, out, 0, N);
    edge_agg_kernel<<<cdiv(E * 32, B), B, 0, stream>>>(rowIdx, colIdx, dinv, H0, out, 0, E);
    bias_relu_kernel<<<cdiv(N * 128, B), B, 0, stream>>>(out, b1, 0, N);

    // ---- layer 2: h2 = relu(agg(h1 @ W2) + b2) -> d_out cols [128,256)
    // A = h1 read from d_out with row stride 256
    gemm_wmma_f32_kernel<<<gemmGrid, 256, 0, stream>>>(out, 256, W2, H0, N);
    self_loop_kernel<<<cdiv(N * 32, B), B, 0, stream>>>(H0, dinv, out, 128, N);
    edge_agg_kernel<<<cdiv(E * 32, B), B, 0, stream>>>(rowIdx, colIdx, dinv, H0, out, 128, E);
    bias_relu_kernel<<<cdiv(N * 128, B), B, 0, stream>>>(out, b2, 128, N);
}